// Tse_Loss_26998164423357
// MI455X (gfx1250) — compile-verified
//
#include <hip/hip_runtime.h>
#include <cmath>

// Problem constants (from the reference): B=32, L=256, T=4096, W=128, y_max=129
#define B_    32
#define L_    256
#define T_    4096
#define NW_   128        // n_words = y_max - 1
#define YMAX_ 129

typedef float v2f __attribute__((ext_vector_type(2)));
typedef float v8f __attribute__((ext_vector_type(8)));

// Per-batch metadata layout in d_ws (ints):
// [0]=n_valid, st @8 (128), ed @136 (128), beg_frame @264 (128), end_frame @392 (128)
#define MSTRIDE 520

__device__ __forceinline__ int sub4i(int y) { return ((y - 1) / 2 - 1) / 2; }

// Fast softplus(-|x|) = log(1 + exp(-|x|)) using v_exp_f32 / v_log_f32.
__device__ __forceinline__ float softplus_negabs(float x) {
  return __logf(1.0f + __expf(-fabsf(x)));
}

// ---------------------------------------------------------------------------
// Metadata: stream-compact valid begin/end token positions (order-preserving)
// and their sub4 frame values. One block per batch, LDS inclusive scan.
// ---------------------------------------------------------------------------
__global__ __launch_bounds__(256) void meta_kernel(const int* __restrict__ wbg,
                                                   const int* __restrict__ wen,
                                                   int* __restrict__ meta) {
  const int b = blockIdx.x, tid = threadIdx.x;
  __shared__ int sb[256], se[256];
  const int vb = wbg[b * L_ + tid];
  const int ve = wen[b * L_ + tid];
  const int fb = (vb != -1) ? 1 : 0;
  const int fe = (ve != -1) ? 1 : 0;
  sb[tid] = fb; se[tid] = fe;
  __syncthreads();
  for (int off = 1; off < 256; off <<= 1) {
    const int xb = (tid >= off) ? sb[tid - off] : 0;
    const int xe = (tid >= off) ? se[tid - off] : 0;
    __syncthreads();
    sb[tid] += xb; se[tid] += xe;
    __syncthreads();
  }
  int* m = meta + b * MSTRIDE;
  if (fb) {
    const int i = sb[tid] - 1;                 // order-preserved compact index
    if (i < NW_) { m[8 + i] = tid; m[8 + 256 + i] = sub4i(vb); }
  }
  if (fe) {
    const int i = se[tid] - 1;
    if (i < NW_) { m[8 + 128 + i] = tid + 1; m[8 + 384 + i] = sub4i(ve); }
  }
  if (tid == 255) m[0] = (sb[255] < NW_) ? sb[255] : NW_;   // n_valid
}

// ---------------------------------------------------------------------------
// Main fused kernel: sparse-window WMMA einsum + target + mask + BCE + reduce.
// Grid: b (32) x row-tile j (8) x t-tile (32 of width 128). 256 thr = 8 waves,
// each wave computes one 16x16 tile of word_mat via v_wmma_f32_16x16x4_f32.
// ---------------------------------------------------------------------------
__global__ __launch_bounds__(256) void tse_main(const float* __restrict__ tse,
                                                const int* __restrict__ meta,
                                                const int* __restrict__ enc_len,
                                                float* __restrict__ out) {
  const int tid = threadIdx.x;
  int bid = blockIdx.x;
  const int tt = bid & 31; bid >>= 5;      // t-tile (0..31), width 128
  const int j  = bid & 7;  bid >>= 3;      // row tile (0..7), 16 word-rows
  const int b  = bid;                      // batch (0..31)
  const int t0 = tt * 128;

  __shared__ int s_st[NW_], s_ed[NW_];
  __shared__ int s_tf[NW_];                // beg_frame | (end_frame << 16)
  __shared__ float red[256];
  __shared__ int s_nv, s_el;

  const int* m = meta + b * MSTRIDE;
  if (tid < 128) {
    s_st[tid] = m[8 + tid];
    s_ed[tid] = m[8 + 128 + tid];
    s_tf[tid] = m[8 + 256 + tid] | (m[8 + 384 + tid] << 16);
  }
  if (tid == 0) { s_nv = m[0]; s_el = enc_len[b]; }
  __syncthreads();
  const int nv = s_nv, el = s_el;

  const int lane = tid & 31, wid = tid >> 5;
  const int tn0  = t0 + wid * 16;          // this wave's 16-column sub-tile
  const int w0   = j * 16;                 // first word index of this row tile
  float sum = 0.f;

  if (w0 < nv) {                           // block-uniform branch
    const int hi   = lane >> 4;            // 0: K+0/K+1 half, 1: K+2/K+3 half
    const int myW  = w0 + (lane & 15);     // A-matrix row M = lane % 16
    const bool mval = (myW < nv);
    // span test as one unsigned range compare: st < k <= ed
    //   <=>  (unsigned)(k - (st+1)) < (unsigned)(ed - st)
    int myst1 = 0, mylen = 0;
    if (mval) {
      const int st = s_st[myW];
      int ed = s_ed[myW]; if (ed > L_ - 1) ed = L_ - 1;
      myst1 = st + 1;
      mylen = ed - st; if (mylen < 0) mylen = 0;
    }

    const int wlast = (w0 + 15 < nv - 1) ? (w0 + 15) : (nv - 1);
    const int klo = (s_st[w0] + 1) & ~3;                       // aligned down
    int khi = s_ed[wlast]; if (khi > L_ - 1) khi = L_ - 1;     // last token

    const int tcol = tn0 + (lane & 15);    // B/D column N = lane % 16
    // Base pointer already folded with this lane's k-phase (2*hi rows).
    const float* kp = tse + (size_t)b * L_ * T_ + tcol + (size_t)(klo + 2 * hi) * T_;

    v8f c = {0.f, 0.f, 0.f, 0.f, 0.f, 0.f, 0.f, 0.f};
    v2f bcur;                              // B fragment, 1-deep pipeline
    bcur.x = kp[0];
    bcur.y = kp[T_];

    int k = klo;
    for (; k + 4 <= khi; k += 4) {         // steady state: prefetch next block
      kp += 4 * T_;                        // constant stride: 0x10000 bytes
      v2f bnext;
      bnext.x = kp[0];
      bnext.y = kp[T_];
      const int kx = k + 2 * hi;
      v2f a;
      a.x = ((unsigned)(kx     - myst1) < (unsigned)mylen) ? 1.f : 0.f;
      a.y = ((unsigned)(kx + 1 - myst1) < (unsigned)mylen) ? 1.f : 0.f;
      c = __builtin_amdgcn_wmma_f32_16x16x4_f32(
              false, a, false, bcur, (short)0, c, false, false);
      bcur = bnext;
    }
    {                                      // peeled final k-block
      const int kx = k + 2 * hi;
      v2f a;
      a.x = ((unsigned)(kx     - myst1) < (unsigned)mylen) ? 1.f : 0.f;
      a.y = ((unsigned)(kx + 1 - myst1) < (unsigned)mylen) ? 1.f : 0.f;
      c = __builtin_amdgcn_wmma_f32_16x16x4_f32(
              false, a, false, bcur, (short)0, c, false, false);
    }

    // Branchless epilogue. D layout: VGPR r holds M=r (lanes 0-15) / M=r+8
    // (lanes 16-31). Packed frame bounds -> one ds_load per accumulator row.
    const int colok = (tcol < el) ? 1 : 0;
#pragma unroll
    for (int r = 0; r < 8; ++r) {
      const int yw    = w0 + r + 8 * hi;     // word index of this acc row
      const int rowok = (yw < nv) ? 1 : 0;
      const int pk    = s_tf[rowok ? yw : (NW_ - 1)];
      const int bf    = pk & 0xffff;
      const int ef    = pk >> 16;
      const float x   = c[r];
      const float xt  = ((tcol >= bf) & (tcol < ef)) ? x : 0.f;  // x * target
      const float bce = fmaxf(x, 0.f) - xt + softplus_negabs(x);
      sum += (rowok & colok) ? bce : 0.f;
    }
  }

  // Row 0 ("no-word" row): handled once per t-tile by the j==0 blocks.
  if (j == 0 && tid < 128) {
    const int t = t0 + tid;
    if (t < el) {
      const float x = tse[(size_t)b * L_ * T_ + t];   // token l = 0
      int cov = 0;
      for (int w = 0; w < nv; ++w) {
        const int pk = s_tf[w];
        cov += ((t >= (pk & 0xffff)) & (t < (pk >> 16))) ? 1 : 0;
      }
      const float tgt0 = fmaxf(0.f, 1.f - (float)cov);
      sum += fmaxf(x, 0.f) - x * tgt0 + softplus_negabs(x);
    }
  }

  // Block reduction + one atomic per block (mean scaling folded in).
  red[tid] = sum;
  __syncthreads();
  for (int off = 128; off > 0; off >>= 1) {
    if (tid < off) red[tid] += red[tid + off];
    __syncthreads();
  }
  if (tid == 0) {
    const float invN = 1.0f / ((float)B_ * (float)YMAX_ * (float)T_);
    atomicAdd(out, red[0] * invN);
  }
}

extern "C" void kernel_launch(void* const* d_in, const int* in_sizes, int n_in,
                              void* d_out, int out_size, void* d_ws, size_t ws_size,
                              hipStream_t stream) {
  (void)in_sizes; (void)n_in; (void)out_size; (void)ws_size;
  const float* tse = (const float*)d_in[0];
  const int*   wbg = (const int*)d_in[1];
  const int*   wen = (const int*)d_in[2];
  const int*   enc = (const int*)d_in[3];
  int* meta = (int*)d_ws;                       // 32 * 520 ints = 66.6 KB

  hipMemsetAsync(d_out, 0, sizeof(float), stream);   // harness poisons d_out
  meta_kernel<<<B_, 256, 0, stream>>>(wbg, wen, meta);
  tse_main<<<B_ * 8 * 32, 256, 0, stream>>>(tse, meta, enc, (float*)d_out);
}